// LstmRNN_12206297055796
// MI455X (gfx1250) — compile-verified
//
#include <hip/hip_runtime.h>
#include <stdint.h>

// ---------------- problem constants (from reference) ----------------
#define SEQ    512
#define BATCH  8192
#define IN     2
#define HID    5
#define GATES  (4 * HID)   // 20, PyTorch gate order: i, f, g, o
#define NPAIR  (GATES / 2) // 10 gate pairs for v_pk_fma_f32
#define BLOCK  64          // 2 wave32s per block -> 128 blocks, 256 waves total
#define DEPTH  16          // async-to-LDS prefetch pipeline depth (steps in flight)

typedef float v2f __attribute__((ext_vector_type(2)));

static __device__ __forceinline__ v2f splat2(float s) {
  v2f r; r.x = s; r.y = s; return r;
}
static __device__ __forceinline__ v2f fma2(v2f a, v2f b, v2f c) {
#if __has_builtin(__builtin_elementwise_fma)
  return __builtin_elementwise_fma(a, b, c);   // -> v_pk_fma_f32
#else
  v2f r; r.x = fmaf(a.x, b.x, c.x); r.y = fmaf(a.y, b.y, c.y); return r;
#endif
}

// ---------------- fast math helpers (gfx1250 trans ops) ----------------
__device__ __forceinline__ float fast_exp2(float x) {
#if __has_builtin(__builtin_amdgcn_exp2f)
  return __builtin_amdgcn_exp2f(x);            // v_exp_f32
#else
  return exp2f(x);
#endif
}
__device__ __forceinline__ float fast_rcp(float x) {
#if __has_builtin(__builtin_amdgcn_rcpf)
  return __builtin_amdgcn_rcpf(x);             // v_rcp_f32
#else
  return 1.0f / x;
#endif
}
__device__ __forceinline__ float ftanh(float x) {
#if __has_builtin(__builtin_amdgcn_tanhf)
  return __builtin_amdgcn_tanhf(x);            // gfx1250 v_tanh_f32
#else
  x = fminf(fmaxf(x, -10.0f), 10.0f);          // avoid inf/inf -> NaN
  float e = fast_exp2(2.88539008177792681472f * x);   // exp(2x)
  return (e - 1.0f) * fast_rcp(e + 1.0f);
#endif
}
__device__ __forceinline__ float fsigmoid(float x) {
#if __has_builtin(__builtin_amdgcn_tanhf)
  return fmaf(0.5f, __builtin_amdgcn_tanhf(0.5f * x), 0.5f);   // 3 VALU ops
#else
  return fast_rcp(1.0f + fast_exp2(-1.44269504088896340736f * x));
#endif
}

// ---------------- CDNA5 async global -> LDS copy (8 bytes / lane) ----------------
// GVS mode: global addr = SADDR(64b) + VADDR(32b byte offset); LDS dest addr from
// VDST VGPR. Tracked by ASYNCcnt; in-order completion -> sliding-window wait works.
__device__ __forceinline__ void async_load_x8(unsigned lds_addr, const float* base,
                                              unsigned voff) {
  asm volatile("global_load_async_to_lds_b64 %0, %1, %2"
               :
               : "v"(lds_addr), "v"(voff), "s"(base)
               : "memory");
}

__global__ __launch_bounds__(BLOCK)
void lstm_fused_kernel(const float* __restrict__ x,      // [SEQ, BATCH, IN]
                       const float* __restrict__ W_ih,   // [GATES, IN]
                       const float* __restrict__ W_hh,   // [GATES, HID]
                       const float* __restrict__ b_ih,   // [GATES]
                       const float* __restrict__ b_hh,   // [GATES]
                       const float* __restrict__ W_lin,  // [1, HID]
                       const float* __restrict__ b_lin,  // [1]
                       float* __restrict__ out)          // [BATCH, 1]
{
  // Per-thread prefetch FIFO: DEPTH steps of this thread's (x0,x1) pair.
  __shared__ float2 fifo[DEPTH][BLOCK];
  // Weight staging area: routes uniform weights through LDS so every thread
  // re-reads them as *vector* (per-lane VGPR) values -> no SGPR spills /
  // v_readlane traffic inside the 512-step sequential loop.
  __shared__ v2f s_wih0[NPAIR], s_wih1[NPAIR], s_bsum[NPAIR], s_whh[HID][NPAIR];
  __shared__ float s_wlin[HID], s_blin;

  const int tid = (int)threadIdx.x;
  const int b   = (int)blockIdx.x * BLOCK + tid;   // batch element owned by this thread

  // ---- one thread stages packed weights into LDS (one-time cost) ----
  if (tid == 0) {
#pragma unroll
    for (int j = 0; j < NPAIR; ++j) {
      const int g0 = 2 * j, g1 = 2 * j + 1;
      v2f t;
      t.x = W_ih[g0 * IN + 0]; t.y = W_ih[g1 * IN + 0]; s_wih0[j] = t;
      t.x = W_ih[g0 * IN + 1]; t.y = W_ih[g1 * IN + 1]; s_wih1[j] = t;
      t.x = b_ih[g0] + b_hh[g0]; t.y = b_ih[g1] + b_hh[g1]; s_bsum[j] = t;
#pragma unroll
      for (int k = 0; k < HID; ++k) {
        t.x = W_hh[g0 * HID + k]; t.y = W_hh[g1 * HID + k]; s_whh[k][j] = t;
      }
    }
#pragma unroll
    for (int k = 0; k < HID; ++k) s_wlin[k] = W_lin[k];
    s_blin = b_lin[0];
  }
  __syncthreads();   // s_barrier_signal/_wait split barrier

  // ---- pull weights into per-lane registers (broadcast LDS reads) ----
  v2f wih0[NPAIR], wih1[NPAIR], bsum[NPAIR], whh[HID][NPAIR];
#pragma unroll
  for (int j = 0; j < NPAIR; ++j) {
    wih0[j] = s_wih0[j]; wih1[j] = s_wih1[j]; bsum[j] = s_bsum[j];
#pragma unroll
    for (int k = 0; k < HID; ++k) whh[k][j] = s_whh[k][j];
  }
  float wlin[HID];
#pragma unroll
  for (int k = 0; k < HID; ++k) wlin[k] = s_wlin[k];
  const float blin = s_blin;

  // ---- async pipeline prologue: DEPTH steps in flight ----
  // Low 32 bits of an LDS flat pointer == workgroup-relative LDS byte offset,
  // exactly what global_load_async_to_lds_b64's VDST operand wants.
  const unsigned lds_base    = (unsigned)(uintptr_t)&fifo[0][tid];
  const unsigned slot_stride = (unsigned)(sizeof(float2) * BLOCK);           // 512 B
  const unsigned gbyte       = (unsigned)b * (unsigned)(IN * sizeof(float)); // x[0][b][0]
  const unsigned step_bytes  = (unsigned)(BATCH * IN * sizeof(float));       // 64 KiB

#pragma unroll
  for (int j = 0; j < DEPTH; ++j)
    async_load_x8(lds_base + (unsigned)j * slot_stride, x,
                  gbyte + (unsigned)j * step_bytes);

  float h[HID] = {0.f, 0.f, 0.f, 0.f, 0.f};
  float c[HID] = {0.f, 0.f, 0.f, 0.f, 0.f};

  auto step = [&](float x0, float x1) {
    // 10 packed gate-pair accumulators -> v_pk_fma_f32 stream (~70 pk-FMAs)
    v2f pre[NPAIR];
#pragma unroll
    for (int j = 0; j < NPAIR; ++j) {
      v2f a = fma2(wih0[j], splat2(x0), bsum[j]);
      a = fma2(wih1[j], splat2(x1), a);
#pragma unroll
      for (int k = 0; k < HID; ++k)
        a = fma2(whh[k][j], splat2(h[k]), a);
      pre[j] = a;
    }
    // unpack (register renaming only; fully unrolled)
    float pg[GATES];
#pragma unroll
    for (int j = 0; j < NPAIR; ++j) { pg[2 * j] = pre[j].x; pg[2 * j + 1] = pre[j].y; }
#pragma unroll
    for (int k = 0; k < HID; ++k) {
      float ig = fsigmoid(pg[0 * HID + k]);
      float fg = fsigmoid(pg[1 * HID + k]);
      float gg = ftanh   (pg[2 * HID + k]);
      float og = fsigmoid(pg[3 * HID + k]);
      c[k] = fmaf(fg, c[k], ig * gg);
      h[k] = og * ftanh(c[k]);
    }
  };

  // ---- main loop: wait oldest slot, consume, refill, compute ----
  for (int s = 0; s < SEQ - DEPTH; ++s) {
    // Oldest of the DEPTH in-flight async loads has landed in LDS.
    asm volatile("s_wait_asynccnt 15" ::: "memory");   // DEPTH-1
    float2 xv = fifo[s & (DEPTH - 1)][tid];            // ds_load_b64
    // WAR guard: slot read must retire before the async refill overwrites it
    // (ASYNC ops are unordered vs DS ops).
    asm volatile("s_wait_dscnt 0x0" ::: "memory");
    async_load_x8(lds_base + (unsigned)(s & (DEPTH - 1)) * slot_stride, x,
                  gbyte + (unsigned)(s + DEPTH) * step_bytes);
    step(xv.x, xv.y);
  }

  // ---- tail: drain remaining in-flight loads, no more refills ----
  asm volatile("s_wait_asynccnt 0x0" ::: "memory");
  for (int s = SEQ - DEPTH; s < SEQ; ++s) {
    float2 xv = fifo[s & (DEPTH - 1)][tid];
    step(xv.x, xv.y);
  }

  // ---- final linear: out[b] = h_last . W_lin + b_lin ----
  float acc = blin;
#pragma unroll
  for (int k = 0; k < HID; ++k) acc = fmaf(h[k], wlin[k], acc);
  out[b] = acc;
}

extern "C" void kernel_launch(void* const* d_in, const int* in_sizes, int n_in,
                              void* d_out, int out_size, void* d_ws, size_t ws_size,
                              hipStream_t stream) {
  (void)in_sizes; (void)n_in; (void)out_size; (void)d_ws; (void)ws_size;
  const float* x     = (const float*)d_in[0];
  const float* W_ih  = (const float*)d_in[1];
  const float* W_hh  = (const float*)d_in[2];
  const float* b_ih  = (const float*)d_in[3];
  const float* b_hh  = (const float*)d_in[4];
  const float* W_lin = (const float*)d_in[5];
  const float* b_lin = (const float*)d_in[6];
  float* out = (float*)d_out;

  dim3 grid(BATCH / BLOCK), block(BLOCK);
  lstm_fused_kernel<<<grid, block, 0, stream>>>(x, W_ih, W_hh, b_ih, b_hh,
                                                W_lin, b_lin, out);
}